// MyModel_61933428414492
// MI455X (gfx1250) — compile-verified
//
#include <hip/hip_runtime.h>

// out(32, N) = sparse-COO W(32x64, nnz) * x(64, N)
// K-compacted WMMA: one K-slot per nonzero.
//   A[m][k] = values[k] * (rows[k]==m)     (16x4 f32, lo/hi halves for 32 rows)
//   B[k][n] = x[cols[k], tile_base + n]    (4x16 f32)
//   D = A*B via v_wmma_f32_16x16x4_f32 (inline-0 accumulator)
// Reads only the nnz touched rows of x -> 140MB traffic floor (12MB in, 128MB out).

typedef __attribute__((ext_vector_type(2))) float v2f;
typedef __attribute__((ext_vector_type(8))) float v8f;

#define TILES_PER_WAVE 4

// ---------------- fast path: nnz <= 4 (single K chunk, setup hoisted) -------
__global__ __launch_bounds__(256) void coo_spmm_wmma_fast(
    const float* __restrict__ x,      // (64, N)
    const float* __restrict__ values, // (nnz)
    const int*   __restrict__ rows,
    const int*   __restrict__ cols,
    float*       __restrict__ out,    // (32, N)
    int nnz, int ncols, int ntiles)
{
    const int lane  = threadIdx.x & 31;
    const int wave  = (blockIdx.x * blockDim.x + threadIdx.x) >> 5;
    const int tile0 = wave * TILES_PER_WAVE;
    if (tile0 >= ntiles) return;                       // wave-uniform

    const int m    = lane & 15;
    const int half = lane >> 4;

    // ---- per-wave setup (divergent gathers reconverge before the loops) ----
    const int k0  = 2 * half;            // this lane's two K slots
    const int k1  = k0 + 1;
    const int kk0 = (k0 < nnz) ? k0 : 0;
    const int kk1 = (k1 < nnz) ? k1 : 0;

    const int   col0 = cols[kk0];
    const int   col1 = cols[kk1];
    const int   r0   = (k0 < nnz) ? rows[kk0]   : -1;
    const int   r1   = (k1 < nnz) ? rows[kk1]   : -1;
    const float v0   = (k0 < nnz) ? values[kk0] : 0.0f;
    const float v1   = (k1 < nnz) ? values[kk1] : 0.0f;

    v2f a_lo, a_hi;
    a_lo.x = (r0 == m)      ? v0 : 0.0f;
    a_lo.y = (r1 == m)      ? v1 : 0.0f;
    a_hi.x = (r0 == m + 16) ? v0 : 0.0f;
    a_hi.y = (r1 == m + 16) ? v1 : 0.0f;

    const int rlo   = 8 * half;          // D row base for this lane half
    const int base0 = tile0 * 16;
    const int nidx0 = base0 + m;

    const bool full = (tile0 + TILES_PER_WAVE <= ntiles) &&
                      (base0 + TILES_PER_WAVE * 16 <= ncols);   // wave-uniform

    if (full) {
        // All addresses hoisted: per-tile offsets are compile-time literals
        // that fold into the 24-bit instruction immediate.
        const float* px0 = x + col0 * ncols + nidx0;
        const float* px1 = x + col1 * ncols + nidx0;

        float* po_lo[8];
        float* po_hi[8];
        #pragma unroll
        for (int i = 0; i < 8; ++i) {
            po_lo[i] = out + (rlo + i)      * ncols + nidx0;
            po_hi[i] = out + (rlo + i + 16) * ncols + nidx0;
        }

        // Batch all loads first (one clause, overlapped latency).
        v2f b[TILES_PER_WAVE];
        #pragma unroll
        for (int t = 0; t < TILES_PER_WAVE; ++t) {
            b[t].x = __builtin_nontemporal_load(px0 + t * 16);
            b[t].y = __builtin_nontemporal_load(px1 + t * 16);
        }

        #pragma unroll
        for (int t = 0; t < TILES_PER_WAVE; ++t) {
            v8f c_lo = {}, c_hi = {};
            c_lo = __builtin_amdgcn_wmma_f32_16x16x4_f32(
                false, a_lo, false, b[t], (short)0, c_lo, false, false);
            c_hi = __builtin_amdgcn_wmma_f32_16x16x4_f32(
                false, a_hi, false, b[t], (short)0, c_hi, false, false);
            #pragma unroll
            for (int i = 0; i < 8; ++i) {
                __builtin_nontemporal_store(c_lo[i], po_lo[i] + t * 16);
                __builtin_nontemporal_store(c_hi[i], po_hi[i] + t * 16);
            }
        }
    } else {
        // Tail waves: guarded per-tile path.
        const int tend = min(tile0 + TILES_PER_WAVE, ntiles);
        const int xoff0 = col0 * ncols;
        const int xoff1 = col1 * ncols;
        for (int t = tile0; t < tend; ++t) {
            const int nidx = t * 16 + m;
            const int cl   = (nidx < ncols) ? nidx : (ncols - 1);
            v2f b;
            b.x = __builtin_nontemporal_load(x + xoff0 + cl);
            b.y = __builtin_nontemporal_load(x + xoff1 + cl);
            v8f c_lo = {}, c_hi = {};
            c_lo = __builtin_amdgcn_wmma_f32_16x16x4_f32(
                false, a_lo, false, b, (short)0, c_lo, false, false);
            c_hi = __builtin_amdgcn_wmma_f32_16x16x4_f32(
                false, a_hi, false, b, (short)0, c_hi, false, false);
            if (nidx < ncols) {
                #pragma unroll
                for (int i = 0; i < 8; ++i) {
                    __builtin_nontemporal_store(c_lo[i], out + (rlo + i)      * ncols + nidx);
                    __builtin_nontemporal_store(c_hi[i], out + (rlo + i + 16) * ncols + nidx);
                }
            }
        }
    }
}

// ---------------- generic path: any nnz (chunked K accumulation) ------------
__global__ __launch_bounds__(256) void coo_spmm_wmma_gen(
    const float* __restrict__ x,
    const float* __restrict__ values,
    const int*   __restrict__ rows,
    const int*   __restrict__ cols,
    float*       __restrict__ out,
    int nnz, int ncols, int ntiles)
{
    const int lane = threadIdx.x & 31;
    const int tile = (blockIdx.x * blockDim.x + threadIdx.x) >> 5;
    if (tile >= ntiles) return;

    const int m    = lane & 15;
    const int half = lane >> 4;
    const int nidx = tile * 16 + m;
    const int cl   = (nidx < ncols) ? nidx : (ncols - 1);

    v8f c_lo = {}, c_hi = {};

    for (int chunk = 0; chunk < nnz; chunk += 4) {
        const int k0  = chunk + 2 * half;
        const int k1  = k0 + 1;
        const int kk0 = (k0 < nnz) ? k0 : 0;
        const int kk1 = (k1 < nnz) ? k1 : 0;

        const int   col0 = cols[kk0];
        const int   col1 = cols[kk1];
        const int   r0   = (k0 < nnz) ? rows[kk0]   : -1;
        const int   r1   = (k1 < nnz) ? rows[kk1]   : -1;
        const float v0   = (k0 < nnz) ? values[kk0] : 0.0f;
        const float v1   = (k1 < nnz) ? values[kk1] : 0.0f;

        v2f a_lo, a_hi, b;
        a_lo.x = (r0 == m)      ? v0 : 0.0f;
        a_lo.y = (r1 == m)      ? v1 : 0.0f;
        a_hi.x = (r0 == m + 16) ? v0 : 0.0f;
        a_hi.y = (r1 == m + 16) ? v1 : 0.0f;
        b.x = __builtin_nontemporal_load(x + col0 * ncols + cl);
        b.y = __builtin_nontemporal_load(x + col1 * ncols + cl);

        c_lo = __builtin_amdgcn_wmma_f32_16x16x4_f32(
            false, a_lo, false, b, (short)0, c_lo, false, false);
        c_hi = __builtin_amdgcn_wmma_f32_16x16x4_f32(
            false, a_hi, false, b, (short)0, c_hi, false, false);
    }

    if (nidx < ncols) {
        const int rlo = 8 * half;
        #pragma unroll
        for (int i = 0; i < 8; ++i) {
            __builtin_nontemporal_store(c_lo[i], out + (rlo + i)      * ncols + nidx);
            __builtin_nontemporal_store(c_hi[i], out + (rlo + i + 16) * ncols + nidx);
        }
    }
}

extern "C" void kernel_launch(void* const* d_in, const int* in_sizes, int n_in,
                              void* d_out, int out_size, void* d_ws, size_t ws_size,
                              hipStream_t stream) {
    const float* x      = (const float*)d_in[0];
    const float* values = (const float*)d_in[1];
    const int*   rows   = (const int*)d_in[2];
    const int*   cols   = (const int*)d_in[3];
    float*       out    = (float*)d_out;

    const int IN_FEATURES = 64;
    const int nnz    = in_sizes[1];
    const int ncols  = in_sizes[0] / IN_FEATURES;  // 1,000,000
    const int ntiles = (ncols + 15) / 16;          // 62,500

    const int threads = 256;                       // 8 wave32s per block

    if (nnz <= 4) {
        const int waves  = (ntiles + TILES_PER_WAVE - 1) / TILES_PER_WAVE;
        const int blocks = (waves + (threads / 32) - 1) / (threads / 32);
        coo_spmm_wmma_fast<<<blocks, threads, 0, stream>>>(
            x, values, rows, cols, out, nnz, ncols, ntiles);
    } else {
        const int blocks = (ntiles + (threads / 32) - 1) / (threads / 32);
        coo_spmm_wmma_gen<<<blocks, threads, 0, stream>>>(
            x, values, rows, cols, out, nnz, ncols, ntiles);
    }
}